// MPMesh_24515673325986
// MI455X (gfx1250) — compile-verified
//
#include <hip/hip_runtime.h>
#include <math.h>

// ---------------- problem constants (from reference) ----------------
#define MPI_D   32
#define HH      256
#define WW      256
#define AH      1024          // ATLAS_GH * H
#define AW      2048          // ATLAS_GW * W
#define FP      98            // faces per plane = 7*7*2
#define NTILE   26            // 4 triangles per 16x16x4 WMMA tile; padded even for x2 unroll
#define NSLOT   (NTILE*4)     // 104 slots per plane (98 real, 6 pad)
#define NVERT   2048          // 32*8*8

// ---------------- workspace layout (float offsets) ----------------
#define WS_E    0             // R(9)+T(3)
#define WS_K    12            // K'(9)
#define WS_XY   32            // 2048*2 projected verts
#define WS_COEF 4160          // 32 planes * 26 tiles * 32 lanes * 2 = 53248 (WMMA A layout)
#define WS_U    57408         // 32*98*6 uv affine maps

#define COEF_PLANE (NTILE*64)       // 1664 floats = 6656 B per plane
#define NCHUNK     (COEF_PLANE/4)   // 416 16-byte chunks per plane

typedef float v2f __attribute__((ext_vector_type(2)));
typedef float v8f __attribute__((ext_vector_type(8)));

// ---------------- phase 1a: fold extrinsics/intrinsics ----------------
__global__ void k_setup(const float* __restrict__ tarE, const float* __restrict__ refE,
                        const float* __restrict__ tarK, float* __restrict__ ws) {
  if (threadIdx.x != 0 || blockIdx.x != 0) return;
  // rigid inverse of ref_extrin: inv([R t;0 1]) = [R^T, -R^T t]
  float Rr[9], tr[3], Ri[9], ti[3];
  for (int r = 0; r < 3; ++r) {
    for (int c = 0; c < 3; ++c) Rr[r*3+c] = refE[r*4+c];
    tr[r] = refE[r*4+3];
  }
  for (int r = 0; r < 3; ++r)
    for (int c = 0; c < 3; ++c) Ri[r*3+c] = Rr[c*3+r];
  for (int r = 0; r < 3; ++r)
    ti[r] = -(Ri[r*3+0]*tr[0] + Ri[r*3+1]*tr[1] + Ri[r*3+2]*tr[2]);
  // E = tar_extrins[0] @ inv(ref_extrin)  (3x4 part)
  float R[9], T[3];
  for (int r = 0; r < 3; ++r) {
    for (int c = 0; c < 3; ++c)
      R[r*3+c] = tarE[r*4+0]*Ri[0*3+c] + tarE[r*4+1]*Ri[1*3+c] + tarE[r*4+2]*Ri[2*3+c];
    T[r] = tarE[r*4+0]*ti[0] + tarE[r*4+1]*ti[1] + tarE[r*4+2]*ti[2] + tarE[r*4+3];
  }
  for (int i = 0; i < 9; ++i) ws[WS_E + i] = R[i];
  for (int i = 0; i < 3; ++i) ws[WS_E + 9 + i] = T[i];
  // intrin = tar_intrins * scale; +1 / +h/w offsets
  float K[9];
  const float s01 = -2.0f / (float)WW;
  for (int c = 0; c < 3; ++c) {
    K[0*3+c] = tarK[0*3+c] * s01;
    K[1*3+c] = tarK[1*3+c] * s01;
    K[2*3+c] = tarK[2*3+c];
  }
  K[0*3+2] += 1.0f;
  K[1*3+2] += (float)HH / (float)WW;
  for (int i = 0; i < 9; ++i) ws[WS_K + i] = K[i];
}

// ---------------- phase 1b: project verts to NDC xy ----------------
__global__ void k_project(const float* __restrict__ verts, float* __restrict__ ws) {
  int v = blockIdx.x * blockDim.x + threadIdx.x;
  if (v >= NVERT) return;
  const float* R = ws + WS_E;
  const float* T = ws + WS_E + 9;
  const float* K = ws + WS_K;
  float x = verts[v*3+0], y = verts[v*3+1], z = verts[v*3+2];
  float vx = R[0]*x + R[1]*y + R[2]*z + T[0];
  float vy = R[3]*x + R[4]*y + R[5]*z + T[1];
  float vz = R[6]*x + R[7]*y + R[8]*z + T[2];
  float nx = K[0]*vx + K[1]*vy + K[2]*vz;
  float ny = K[3]*vx + K[4]*vy + K[5]*vz;
  float nz = K[6]*vx + K[7]*vy + K[8]*vz;
  ws[WS_XY + v*2 + 0] = nx / nz;
  ws[WS_XY + v*2 + 1] = ny / nz;
}

// ---------------- phase 2: per-(plane,triangle) barycentric functionals ----------------
// Pack into WMMA A layout for V_WMMA_F32_16X16X4_F32 (32-bit A 16x4, ISA 7.12.2):
//   lanes 0-15: VGPR0=K0(cx), VGPR1=K1(cy); lanes 16-31: VGPR0=K2(c1), VGPR1=K3(0)
// 4 triangles per tile, each occupying 4 rows (w0,w1,w2,pad); pad row -> w = -1 (never inside).
__global__ void k_coef(const int* __restrict__ faces, const float* __restrict__ uvs,
                       float* __restrict__ ws) {
  int s = blockIdx.x * blockDim.x + threadIdx.x;
  if (s >= MPI_D * NSLOT) return;
  int plane = s / NSLOT, slot = s % NSLOT;
  int tile = slot / 4, sl = slot % 4;
  float* A = ws + WS_COEF + (size_t)(plane * NTILE + tile) * 64;  // 32 lanes * 2
  float cx[4], cy[4], c1[4];
  for (int r = 0; r < 4; ++r) { cx[r] = 0.0f; cy[r] = 0.0f; c1[r] = -1.0f; }  // pad
  if (slot < FP) {
    const int* f = faces + (size_t)(plane * FP + slot) * 3;
    int v0 = f[0], v1 = f[1], v2 = f[2];
    float ax = ws[WS_XY+v0*2], ay = ws[WS_XY+v0*2+1];
    float bx = ws[WS_XY+v1*2], by = ws[WS_XY+v1*2+1];
    float qx = ws[WS_XY+v2*2], qy = ws[WS_XY+v2*2+1];
    float d = (bx-ax)*(qy-ay) - (by-ay)*(qx-ax);
    if (fabsf(d) < 1e-9f) d = 1e-9f;     // matches jnp.where(|d|<1e-9, 1e-9, d)
    float id = 1.0f / d;
    // w1 = cross(p-a, c-a)/d ; w2 = cross(b-a, p-a)/d ; w0 = 1-w1-w2  (all linear in p)
    float cx1 =  (qy-ay)*id, cy1 = -(qx-ax)*id, c11 = (-ax*(qy-ay) + ay*(qx-ax))*id;
    float cx2 = -(by-ay)*id, cy2 =  (bx-ax)*id, c12 = ( ax*(by-ay) - ay*(bx-ax))*id;
    float cx0 = -cx1-cx2,    cy0 = -cy1-cy2,    c10 = 1.0f - c11 - c12;
    cx[0]=cx0; cy[0]=cy0; c1[0]=c10;
    cx[1]=cx1; cy[1]=cy1; c1[1]=c11;
    cx[2]=cx2; cy[2]=cy2; c1[2]=c12;
    // uv affine map: uv = sum_k w_k(p) * uvs[vid_k]  ->  2x3 matrix on [px,py,1]
    float u0 = uvs[v0*2], w0 = uvs[v0*2+1];
    float u1 = uvs[v1*2], w1 = uvs[v1*2+1];
    float u2 = uvs[v2*2], w2 = uvs[v2*2+1];
    float* Ud = ws + WS_U + (size_t)(plane * FP + slot) * 6;
    Ud[0] = cx0*u0 + cx1*u1 + cx2*u2;
    Ud[1] = cy0*u0 + cy1*u1 + cy2*u2;
    Ud[2] = c10*u0 + c11*u1 + c12*u2;
    Ud[3] = cx0*w0 + cx1*w1 + cx2*w2;
    Ud[4] = cy0*w0 + cy1*w1 + cy2*w2;
    Ud[5] = c10*w0 + c11*w1 + c12*w2;
  }
  int rowBase = sl * 4;
  for (int r = 0; r < 4; ++r) {
    int m = rowBase + r;
    A[m*2 + 0]        = cx[r];   // K0
    A[m*2 + 1]        = cy[r];   // K1
    A[(m+16)*2 + 0]   = c1[r];   // K2 (multiplied by constant 1)
    A[(m+16)*2 + 1]   = 0.0f;    // K3
  }
}

// ---------------- async stage: one plane's coef tiles -> LDS buffer ----------------
__device__ __forceinline__ void stage_plane_async(const float* __restrict__ ws, int d,
                                                  unsigned ldsDst, int tid) {
  const float* src = ws + WS_COEF + (size_t)d * COEF_PLANE;
  unsigned long long ga0 = (unsigned long long)(const void*)(src + tid * 4);
  unsigned l0 = ldsDst + (unsigned)(tid * 16);
  asm volatile("global_load_async_to_lds_b128 %0, %1, off" :: "v"(l0), "v"(ga0) : "memory");
  int t1 = tid + 256;
  if (t1 < NCHUNK) {
    unsigned long long ga1 = (unsigned long long)(const void*)(src + t1 * 4);
    unsigned l1 = ldsDst + (unsigned)(t1 * 16);
    asm volatile("global_load_async_to_lds_b128 %0, %1, off" :: "v"(l1), "v"(ga1) : "memory");
  }
}

// ---------------- phase 3: render (search via WMMA + sample + composite) ----------------
__global__ void __launch_bounds__(256) k_render(const float* __restrict__ atlas,
                                                const float* __restrict__ ws,
                                                float* __restrict__ out) {
  __shared__ float ldsCoef[2][COEF_PLANE];   // double buffer: 2 x 6656 B
  const int tid  = threadIdx.x;
  const int lane = tid & 31;
  const int wave = tid >> 5;
  const bool hi  = (lane & 16) != 0;
  const int  pl  = lane & 15;
  const int  pix = blockIdx.x * 128 + wave * 16 + pl;   // 8 waves x 16 pixels
  const int  x = pix & (WW - 1), y = pix >> 8;
  const float px = 1.0f - (2.0f * (float)x + 1.0f) / (float)WW;
  const float py = 1.0f - (2.0f * (float)y + 1.0f) / (float)HH;
  // B operand (4x16 fp32): column = pixel; K-rows = [px, py, 1, 0]
  // layout mirrors A: lanes 0-15 -> K0/K1, lanes 16-31 -> K2/K3
  v2f bvec;
  bvec.x = hi ? 1.0f : px;
  bvec.y = hi ? 0.0f : py;
  const int c0 = hi ? 2 : 0;             // this lane samples atlas channels c0, c0+1
  float accR = 0.0f, accG = 0.0f, accB = 0.0f, trans = 1.0f;
  const unsigned ldsBase0 = (unsigned)(unsigned long long)(&ldsCoef[0][0]);
  const unsigned ldsBase1 = (unsigned)(unsigned long long)(&ldsCoef[1][0]);

  // prologue: stage plane 31 into buffer 0
  stage_plane_async(ws, MPI_D - 1, ldsBase0, tid);
  asm volatile("s_wait_asynccnt 0x0" ::: "memory");
  __syncthreads();

  int cur = 0;
  for (int d = MPI_D - 1; d >= 0; --d) {            // front (near) to back (far)
    // overlap: kick off DMA of next plane into the other buffer
    if (d > 0)
      stage_plane_async(ws, d - 1, (cur == 0) ? ldsBase1 : ldsBase0, tid);
    const float* lc = &ldsCoef[cur][0];

    // ---- triangle search: 13 WMMA pairs evaluate 98(+6 pad) triangles x 16 pixels ----
    int best = -1;
    for (int t = 0; t < NTILE; t += 2) {
      v2f a0 = *(const v2f*)&lc[(t * 32 + lane) * 2];         // ds_load_b64
      v2f a1 = *(const v2f*)&lc[((t + 1) * 32 + lane) * 2];   // ds_load_b64
      v8f cz = {};
      v8f w0 = __builtin_amdgcn_wmma_f32_16x16x4_f32(
          false, a0, false, bvec, (short)0, cz, false, false);
      v8f w1 = __builtin_amdgcn_wmma_f32_16x16x4_f32(
          false, a1, false, bvec, (short)0, cz, false, false);
      // D layout: lane p<16 holds rows 0-7 for pixel p; lane p+16 holds rows 8-15.
      // rows {0-2,4-6} = tris 4t+{0,1} (low half) / rows {8-10,12-14} = 4t+{2,3} (high half)
      const float eps = -1e-6f;
      bool i0 = (w0[0] > eps) && (w0[1] > eps) && (w0[2] > eps);
      bool i1 = (w0[4] > eps) && (w0[5] > eps) && (w0[6] > eps);
      bool i2 = (w1[0] > eps) && (w1[1] > eps) && (w1[2] > eps);
      bool i3 = (w1[4] > eps) && (w1[5] > eps) && (w1[6] > eps);
      int ta = t * 4 + (hi ? 2 : 0);        // candidates ascend within the lane
      best = (best < 0 && i0) ? ta         : best;  // first-hit == argmax(inside)
      best = (best < 0 && i1) ? ta + 1     : best;
      best = (best < 0 && i2) ? ta + 4     : best;
      best = (best < 0 && i3) ? ta + 5     : best;
      if (__ballot(best < 0) == 0) break;   // uniform early exit, EXEC stays full
    }
    int ob   = __shfl_xor(best, 16, 32);    // merge lane halves
    int fidx = (best < 0) ? ob : ((ob < 0) ? best : min(best, ob));
    bool valid = fidx >= 0;
    int  ftri  = valid ? fidx : 0;

    // ---- uv via precomputed affine map ----
    const float* U = ws + WS_U + (size_t)(d * FP + ftri) * 6;
    float uu = U[0]*px + U[1]*py + U[2];
    float vv = U[3]*px + U[4]*py + U[5];

    // ---- bilinear grid-sample (zero pad OOB), 2 channels per lane ----
    float ix = ((uu + 1.0f) * (float)AW - 1.0f) * 0.5f;
    float iy = ((vv + 1.0f) * (float)AH - 1.0f) * 0.5f;
    float x0f = floorf(ix), y0f = floorf(iy);
    float tx = ix - x0f, ty = iy - y0f;
    int x0 = (int)x0f, y0 = (int)y0f;
    float s0 = 0.0f, s1 = 0.0f;
#pragma unroll
    for (int j = 0; j < 2; ++j)
#pragma unroll
      for (int i = 0; i < 2; ++i) {
        int xi = x0 + i, yi = y0 + j;
        bool inb = (xi >= 0) && (xi < AW) && (yi >= 0) && (yi < AH);
        int xc = min(max(xi, 0), AW - 1);
        int yc = min(max(yi, 0), AH - 1);
        float wgt = (i ? tx : 1.0f - tx) * (j ? ty : 1.0f - ty);
        size_t base = (size_t)yc * AW + (size_t)xc;
        float p0 = atlas[(size_t)c0       * (AH * (size_t)AW) + base];
        float p1 = atlas[(size_t)(c0 + 1) * (AH * (size_t)AW) + base];
        s0 += (inb ? p0 : 0.0f) * wgt;
        s1 += (inb ? p1 : 0.0f) * wgt;
      }
    s0 = 1.0f / (1.0f + __expf(-s0));
    s1 = 1.0f / (1.0f + __expf(-s1));
    s0 = valid ? s0 : 0.0f;
    s1 = valid ? s1 : 0.0f;

    // ---- gather 4 channels to low lanes, over-composite ----
    float o0 = __shfl_xor(s0, 16, 32);               // low lanes: b ; high lanes: r
    float o1 = __shfl_xor(s1, 16, 32);               // low lanes: a ; high lanes: g
    float alpha = o1;                                 // meaningful in low lanes
    float wgt = alpha * trans;
    accR += wgt * s0;
    accG += wgt * s1;
    accB += wgt * o0;
    trans *= (1.0f - alpha);

    // next plane's DMA must be complete and all waves past their reads of buf[cur]
    asm volatile("s_wait_asynccnt 0x0" ::: "memory");
    __syncthreads();
    cur ^= 1;
  }

  if (!hi) {
    out[pix * 3 + 0] = accR;
    out[pix * 3 + 1] = accG;
    out[pix * 3 + 2] = accB;
  }
}

// ---------------- host entry ----------------
extern "C" void kernel_launch(void* const* d_in, const int* in_sizes, int n_in,
                              void* d_out, int out_size, void* d_ws, size_t ws_size,
                              hipStream_t stream) {
  (void)in_sizes; (void)n_in; (void)out_size; (void)ws_size;
  const float* verts = (const float*)d_in[0];
  const float* uvs   = (const float*)d_in[1];
  const float* atlas = (const float*)d_in[2];
  const float* tarE  = (const float*)d_in[3];
  const float* tarK  = (const float*)d_in[4];
  const float* refE  = (const float*)d_in[5];
  const int*   faces = (const int*)d_in[6];
  float* ws  = (float*)d_ws;
  float* out = (float*)d_out;

  k_setup  <<<1, 1, 0, stream>>>(tarE, refE, tarK, ws);
  k_project<<<(NVERT + 255) / 256, 256, 0, stream>>>(verts, ws);
  k_coef   <<<(MPI_D * NSLOT + 255) / 256, 256, 0, stream>>>(faces, uvs, ws);
  k_render <<<(HH * WW) / 128, 256, 0, stream>>>(atlas, ws, out);
}